// MoE_for_Pruning_26096221290521
// MI455X (gfx1250) — compile-verified
//
#include <hip/hip_runtime.h>
#include <math.h>

#define BATCH   64
#define N_ENT   50000
#define HID     64
#define N_NODE  300000
#define N_OLD   150000
#define N_EDGE  600000
#define ROWCAP  8192
#define CAPQ    8192
#define NPART   256

typedef float v2f __attribute__((ext_vector_type(2)));
typedef float v8f __attribute__((ext_vector_type(8)));

// monotone float->uint map (total order matches float <)
__device__ __forceinline__ unsigned f2u(float f) {
    unsigned b = __float_as_uint(f);
    return (b & 0x80000000u) ? ~b : (b | 0x80000000u);
}
__device__ __forceinline__ float sigmoidf_(float x) { return 1.0f / (1.0f + expf(-x)); }

// ---------------------------------------------------------------- init
__global__ void k_init(float* f0, float* f1, unsigned char* isnew, int* deg,
                       int* cursor, int* row_count, float* meta) {
    int i = blockIdx.x * blockDim.x + threadIdx.x;
    if (i < N_NODE) {
        f0[i] = 1.0f; f1[i] = 1.0f; isnew[i] = 1; deg[i] = 0; cursor[i] = 0;
    }
    if (i < 64) row_count[i] = 0;
    if (i == 0) ((int*)meta)[8] = 0;   // eq_count
}

__global__ void k_mark_old(unsigned char* isnew, const int* old_idx) {
    int i = blockIdx.x * blockDim.x + threadIdx.x;
    if (i < N_OLD) isnew[old_idx[i]] = 0;
}

// ---------------------------------------------------------------- gating (1 block, 256 thr)
__global__ void k_gating(const float* __restrict__ sub_tab, const float* __restrict__ rel_tab,
                         const float* __restrict__ W1, const float* __restrict__ b1,
                         const float* __restrict__ W2, const float* __restrict__ b2,
                         const float* __restrict__ Wn, const float* __restrict__ noise,
                         const int* __restrict__ q_sub, const int* __restrict__ q_rel,
                         const int* __restrict__ l_ptr, float* meta) {
    __shared__ float comb[BATCH][2 * HID];   // 32 KB
    __shared__ float hidA[BATCH][HID];       // 16 KB
    __shared__ float wmat[BATCH][3];
    int t = threadIdx.x;
    for (int idx = t; idx < BATCH * 2 * HID; idx += 256) {
        int b = idx >> 7, k = idx & 127;
        comb[b][k] = (k < HID) ? sub_tab[(size_t)q_sub[b] * HID + k]
                               : rel_tab[(size_t)q_rel[b] * HID + (k - HID)];
    }
    __syncthreads();
    for (int idx = t; idx < BATCH * HID; idx += 256) {
        int b = idx >> 6, j = idx & 63;
        float acc = b1[j];
        const float* wrow = W1 + (size_t)j * 2 * HID;
        for (int k = 0; k < 2 * HID; ++k) acc += comb[b][k] * wrow[k];
        hidA[b][j] = fmaxf(acc, 0.0f);
    }
    __syncthreads();
    if (t < BATCH) {
        int b = t;
        float z[3];
        for (int c = 0; c < 3; ++c) {
            float acc = b2[c];
            const float* wrow = W2 + (size_t)c * HID;
            for (int j = 0; j < HID; ++j) acc += hidA[b][j] * wrow[j];
            z[c] = acc;
        }
        float m = fmaxf(z[0], fmaxf(z[1], z[2]));
        float e0 = expf(z[0] - m), e1 = expf(z[1] - m), e2 = expf(z[2] - m);
        float es = e0 + e1 + e2;
        float g[3] = {e0 / es, e1 / es, e2 / es};
        float x = 0.0f;
        for (int k = 0; k < 2 * HID; ++k) x += comb[b][k] * Wn[k];
        float scale = log1pf(expf(-fabsf(x))) + fmaxf(x, 0.0f);   // softplus
        float y[3];
        for (int c = 0; c < 3; ++c) y[c] = (g[c] + noise[b * 3 + c] * scale) * 2.0f;
        float ym = fmaxf(y[0], fmaxf(y[1], y[2]));
        float f0e = expf(y[0] - ym), f1e = expf(y[1] - ym), f2e = expf(y[2] - ym);
        float fs = f0e + f1e + f2e;
        wmat[b][0] = f0e / fs; wmat[b][1] = f1e / fs; wmat[b][2] = f2e / fs;
    }
    __syncthreads();
    if (t == 0) {
        float cs[3] = {0, 0, 0};
        for (int b = 0; b < BATCH; ++b)
            for (int c = 0; c < 3; ++c) cs[c] += wmat[b][c];
        float tot = cs[0] + cs[1] + cs[2];
        float imp[3] = {cs[0] / tot, cs[1] / tot, cs[2] / tot};
        int i1 = 0;
        for (int c = 1; c < 3; ++c) if (imp[c] > imp[i1]) i1 = c;
        int i2 = -1;
        for (int c = 0; c < 3; ++c) {
            if (c == i1) continue;
            if (i2 < 0 || imp[c] > imp[i2]) i2 = c;
        }
        float s2 = imp[i1] + imp[i2];
        float ew[3] = {0, 0, 0};
        ew[i1] = imp[i1] / s2; ew[i2] = imp[i2] / s2;
        meta[0] = ew[0]; meta[1] = ew[1]; meta[2] = ew[2];
        // K_l from l (compute_topk, double math like Python)
        double lf = (double)l_ptr[0]; double K;
        if (lf < 4.0) { double S = 1.0 / (1.0 + exp(-1.5 * (lf - 2.0))); K = 100.0 + 900.0 * S; }
        else          { double S = 1.0 / (1.0 + exp(-1.5 * (lf - 6.0))); K = 50.0 + 950.0 * (1.0 - S); }
        ((int*)meta)[3] = (int)K;
    }
}

// ---------------------------------------------------------------- candidate scatter
__global__ void k_cand(const int* __restrict__ nodes, const unsigned char* __restrict__ isnew,
                       const float* __restrict__ scores, const float* __restrict__ hidden,
                       const float* __restrict__ rel_tab, const int* __restrict__ q_rel,
                       int* row_count, int* cand_node, int* cand_col,
                       float* cand_s, float* cand_q) {
    int i = blockIdx.x * blockDim.x + threadIdx.x;
    if (i >= N_NODE) return;
    int r = nodes[2 * i], c = nodes[2 * i + 1];
    if (!isnew[i]) return;
    const float* hrow = hidden + (size_t)i * HID;
    const float* rrow = rel_tab + (size_t)q_rel[r] * HID;
    float dot = 0.0f;
    for (int k = 0; k < HID; ++k) dot += hrow[k] * rrow[k];
    int pos = atomicAdd(&row_count[r], 1);
    if (pos < ROWCAP) {
        int idx = r * ROWCAP + pos;
        cand_node[idx] = i; cand_col[idx] = c;
        cand_s[idx] = scores[i]; cand_q[idx] = dot;
    }
}

// ---------------------------------------------------------------- per-row softmax stats
__global__ void k_rowstats(const int* row_count, const float* cand_s, const float* cand_q,
                           float* rowstats) {
    __shared__ float red[256];
    int r = blockIdx.x, t = threadIdx.x;
    int cnt = min(row_count[r], ROWCAP);
    int base = r * ROWCAP;
    float ms = -INFINITY, mq = -INFINITY;
    for (int j = t; j < cnt; j += 256) { ms = fmaxf(ms, cand_s[base + j]); mq = fmaxf(mq, cand_q[base + j]); }
    red[t] = ms; __syncthreads();
    for (int o = 128; o > 0; o >>= 1) { if (t < o) red[t] = fmaxf(red[t], red[t + o]); __syncthreads(); }
    ms = red[0]; __syncthreads();
    red[t] = mq; __syncthreads();
    for (int o = 128; o > 0; o >>= 1) { if (t < o) red[t] = fmaxf(red[t], red[t + o]); __syncthreads(); }
    mq = red[0]; __syncthreads();
    float ss = 0.0f, sq = 0.0f;
    for (int j = t; j < cnt; j += 256) { ss += expf(cand_s[base + j] - ms); sq += expf(cand_q[base + j] - mq); }
    red[t] = ss; __syncthreads();
    for (int o = 128; o > 0; o >>= 1) { if (t < o) red[t] += red[t + o]; __syncthreads(); }
    ss = red[0]; __syncthreads();
    red[t] = sq; __syncthreads();
    for (int o = 128; o > 0; o >>= 1) { if (t < o) red[t] += red[t + o]; __syncthreads(); }
    sq = red[0];
    if (t == 0) { rowstats[r * 4] = ms; rowstats[r * 4 + 1] = mq; rowstats[r * 4 + 2] = ss; rowstats[r * 4 + 3] = sq; }
}

// ---------------------------------------------------------------- per-candidate rank -> f0/f1
__global__ void k_rank(const int* row_count, const int* cand_node, const int* cand_col,
                       const float* cand_s, const float* cand_q, const float* rowstats,
                       const float* meta, float* f0, float* f1) {
    int tid = blockIdx.x * blockDim.x + threadIdx.x;
    if (tid >= 64 * ROWCAP) return;
    int r = tid / ROWCAP, j = tid % ROWCAP;
    int cnt = min(row_count[r], ROWCAP);
    if (j >= cnt) return;
    int base = r * ROWCAP;
    float s = cand_s[base + j], q = cand_q[base + j];
    int col = cand_col[base + j];
    int K = ((const int*)meta)[3];
    int cs = 0, cq = 0;
    for (int k = 0; k < cnt; ++k) {
        float sk = cand_s[base + k], qk = cand_q[base + k];
        int ck = cand_col[base + k];
        cs += (sk > s) || (sk == s && ck < col);
        cq += (qk > q) || (qk == q && ck < col);
    }
    float ms = rowstats[r * 4], mq = rowstats[r * 4 + 1];
    float ss = rowstats[r * 4 + 2], sq = rowstats[r * 4 + 3];
    float ps = expf(s - ms) / ss, pq = expf(q - mq) / sq;
    float hs = (cs < K) ? 1.0f : 0.0f, hq = (cq < K) ? 1.0f : 0.0f;
    int node = cand_node[base + j];
    f0[node] = (hs - ps) + ps;
    f1[node] = (hq - pq) + pq;
}

// ---------------------------------------------------------------- edge softmax reductions
__global__ void k_alpha_partmax(const float* alpha, float* partials) {
    __shared__ float red[256];
    float m = -INFINITY;
    for (int i = blockIdx.x * 256 + threadIdx.x; i < N_EDGE; i += NPART * 256) m = fmaxf(m, alpha[i]);
    red[threadIdx.x] = m; __syncthreads();
    for (int o = 128; o > 0; o >>= 1) { if (threadIdx.x < o) red[threadIdx.x] = fmaxf(red[threadIdx.x], red[threadIdx.x + o]); __syncthreads(); }
    if (threadIdx.x == 0) partials[blockIdx.x] = red[0];
}
__global__ void k_alpha_finmax(const float* partials, float* meta) {
    __shared__ float red[256];
    float m = (threadIdx.x < NPART) ? partials[threadIdx.x] : -INFINITY;
    red[threadIdx.x] = m; __syncthreads();
    for (int o = 128; o > 0; o >>= 1) { if (threadIdx.x < o) red[threadIdx.x] = fmaxf(red[threadIdx.x], red[threadIdx.x + o]); __syncthreads(); }
    if (threadIdx.x == 0) meta[4] = red[0];
}
__global__ void k_alpha_partsum(const float* alpha, const float* meta, float* partials) {
    __shared__ float red[256];
    float M = meta[4], s = 0.0f;
    for (int i = blockIdx.x * 256 + threadIdx.x; i < N_EDGE; i += NPART * 256) s += expf(alpha[i] - M);
    red[threadIdx.x] = s; __syncthreads();
    for (int o = 128; o > 0; o >>= 1) { if (threadIdx.x < o) red[threadIdx.x] += red[threadIdx.x + o]; __syncthreads(); }
    if (threadIdx.x == 0) partials[blockIdx.x] = red[0];
}
__global__ void k_alpha_finsum(const float* partials, float* meta) {
    __shared__ float red[256];
    red[threadIdx.x] = (threadIdx.x < NPART) ? partials[threadIdx.x] : 0.0f; __syncthreads();
    for (int o = 128; o > 0; o >>= 1) { if (threadIdx.x < o) red[threadIdx.x] += red[threadIdx.x + o]; __syncthreads(); }
    if (threadIdx.x == 0) meta[5] = red[0];
}
__global__ void k_ep(const float* alpha, const float* meta, float* ep) {
    int e = blockIdx.x * blockDim.x + threadIdx.x;
    if (e >= N_EDGE) return;
    ep[e] = expf(alpha[e] - meta[4]) / meta[5];
}

// ---------------------------------------------------------------- radix select (top-K of edge_prob)
__global__ void k_hist_zero(unsigned* hist, float* meta, int initflag) {
    int i = blockIdx.x * blockDim.x + threadIdx.x;
    if (i < 2048) hist[i] = 0u;
    if (i == 0 && initflag) { ((unsigned*)meta)[6] = 0u; ((int*)meta)[7] = ((int*)meta)[3]; }
}
__global__ void k_hist(const float* ep, const int* edges, const unsigned char* isnew,
                       unsigned* hist, const float* meta, int s, int hi_shift, int nbins) {
    int e = blockIdx.x * blockDim.x + threadIdx.x;
    if (e >= N_EDGE) return;
    if (!isnew[edges[e * 6 + 5]]) return;
    unsigned u = f2u(ep[e]);
    if (hi_shift < 32 && (u >> hi_shift) != ((const unsigned*)meta)[6]) return;
    atomicAdd(&hist[(u >> s) & (unsigned)(nbins - 1)], 1u);
}
__global__ void k_scan(const unsigned* hist, float* meta, int w, int nbins) {
    unsigned prefix = ((unsigned*)meta)[6];
    int rem = ((int*)meta)[7];
    unsigned cum = 0; int chosen = 0;
    for (int b = nbins - 1; b >= 0; --b) {
        unsigned c = hist[b];
        if (cum + c >= (unsigned)rem) { chosen = b; break; }
        cum += c;
    }
    ((unsigned*)meta)[6] = (prefix << w) | (unsigned)chosen;
    ((int*)meta)[7] = rem - (int)cum;
}

// ---------------------------------------------------------------- ties + edge gates
__global__ void k_eq_compact(const float* ep, const int* edges, const unsigned char* isnew,
                             float* meta, int* eqlist) {
    int e = blockIdx.x * blockDim.x + threadIdx.x;
    if (e >= N_EDGE) return;
    if (!isnew[edges[e * 6 + 5]]) return;
    if (f2u(ep[e]) != ((const unsigned*)meta)[6]) return;
    int pos = atomicAdd(&((int*)meta)[8], 1);
    if (pos < CAPQ) eqlist[pos] = e;
}
__global__ void k_abase(const float* ep, const float* alpha, const int* edges,
                        const unsigned char* isnew, const float* meta, float* ae) {
    int e = blockIdx.x * blockDim.x + threadIdx.x;
    if (e >= N_EDGE) return;
    int newt = isnew[edges[e * 6 + 5]];
    float p = ep[e];
    unsigned thr = ((const unsigned*)meta)[6];
    float hard = (!newt || f2u(p) > thr) ? 1.0f : 0.0f;
    float fct = (hard - p) + p;
    ae[e] = sigmoidf_(alpha[e] * fct);
}
__global__ void k_tiefix(const float* ep, const float* alpha, const float* meta,
                         const int* eqlist, float* ae) {
    int i = blockIdx.x * blockDim.x + threadIdx.x;
    int cnt = min(((const int*)meta)[8], CAPQ);
    if (i >= cnt) return;
    int e = eqlist[i];
    int rank = 0;
    for (int j = 0; j < cnt; ++j) rank += (eqlist[j] < e);
    if (rank < ((const int*)meta)[7]) {
        float p = ep[e];
        float fct = (1.0f - p) + p;
        ae[e] = sigmoidf_(alpha[e] * fct);
    }
}

// ---------------------------------------------------------------- deterministic CSR + segment sum
__global__ void k_csr_count(const int* edges, int* deg) {
    int e = blockIdx.x * blockDim.x + threadIdx.x;
    if (e >= N_EDGE) return;
    atomicAdd(&deg[edges[e * 6 + 5]], 1);
}
__global__ void k_scan_off(const int* deg, int* off) {
    __shared__ int buf[1024];
    __shared__ int carry;
    if (threadIdx.x == 0) carry = 0;
    __syncthreads();
    for (int base = 0; base < N_NODE; base += 1024) {
        int i = base + threadIdx.x;
        int v = (i < N_NODE) ? deg[i] : 0;
        buf[threadIdx.x] = v; __syncthreads();
        for (int o = 1; o < 1024; o <<= 1) {
            int t = (threadIdx.x >= o) ? buf[threadIdx.x - o] : 0;
            __syncthreads();
            buf[threadIdx.x] += t;
            __syncthreads();
        }
        if (i < N_NODE) off[i] = carry + buf[threadIdx.x] - v;
        __syncthreads();
        if (threadIdx.x == 1023) carry += buf[1023];
        __syncthreads();
    }
    if (threadIdx.x == 0) off[N_NODE] = carry;
}
__global__ void k_csr_fill(const int* edges, const int* off, int* cursor, int* elist) {
    int e = blockIdx.x * blockDim.x + threadIdx.x;
    if (e >= N_EDGE) return;
    int obj = edges[e * 6 + 5];
    int pos = off[obj] + atomicAdd(&cursor[obj], 1);
    elist[pos] = e;
}
// 8 threads per node; edges summed in ascending edge-id order -> deterministic
__global__ void k_agg(const int* deg, const int* off, const int* elist,
                      const float* __restrict__ ae, const float* __restrict__ message,
                      float* __restrict__ agg) {
    int tid = blockIdx.x * blockDim.x + threadIdx.x;
    int n = tid >> 3, lane8 = tid & 7;
    if (n >= N_NODE) return;
    int d = deg[n], base = off[n];
    float acc[8] = {0, 0, 0, 0, 0, 0, 0, 0};
    int last = -1;
    for (int t = 0; t < d; ++t) {
        int m = 0x7fffffff;
        for (int j = 0; j < d; ++j) {
            int e = elist[base + j];
            if (e > last && e < m) m = e;
        }
        last = m;
        float a = ae[m];
        const float* mr = message + (size_t)m * HID + lane8 * 8;
        #pragma unroll
        for (int q = 0; q < 8; ++q) acc[q] += a * mr[q];
    }
    float* out = agg + (size_t)n * HID + lane8 * 8;
    #pragma unroll
    for (int q = 0; q < 8; ++q) out[q] = acc[q];
}

// ---------------------------------------------------------------- WMMA GEMM (agg @ W_alpha^T) + combine
// V_WMMA_F32_16X16X4_F32: one wave computes a 16x16 f32 tile, K looped in steps of 4.
// A layout (16x4 f32): lanes 0-15 row M=lane hold K={k0,k0+1}; lanes 16-31 same rows hold K={k0+2,k0+3}.
// B layout (4x16 f32): mirrored; per-lane float2 = W_alpha[n][k0+2*hi .. +1] with n = lane&15.
__global__ void k_gemm_combine(const float* __restrict__ agg, const float* __restrict__ Walpha,
                               const float* __restrict__ hidden, const float* __restrict__ f0,
                               const float* __restrict__ f1, const float* __restrict__ meta,
                               float* __restrict__ hc) {
    int lane = threadIdx.x & 31;
    int wave = threadIdx.x >> 5;
    int tile = blockIdx.x * 8 + wave;          // 75000 tiles total, exact fit
    int tm = tile >> 2, tn = tile & 3;
    int row0 = tm * 16, col0 = tn * 16;
    int hi = lane >> 4, ln = lane & 15;
    int arow = row0 + ln;
    int brow = col0 + ln;
    v8f c = {};
    #pragma unroll
    for (int k0 = 0; k0 < HID; k0 += 4) {
        int k = k0 + 2 * hi;
        v2f a = *(const v2f*)(agg + (size_t)arow * HID + k);
        v2f b = *(const v2f*)(Walpha + (size_t)brow * HID + k);
        c = __builtin_amdgcn_wmma_f32_16x16x4_f32(false, a, false, b, (short)0, c, false, false);
    }
    float w0 = meta[0], w1 = meta[1], w2 = meta[2];
    int col = col0 + ln;
    #pragma unroll
    for (int v = 0; v < 8; ++v) {
        int row = row0 + v + 8 * hi;
        float val = (w0 * f0[row] + w1 * f1[row]) * hidden[(size_t)row * HID + col]
                    + w2 * tanhf(c[v]);
        hc[(size_t)row * HID + col] = val;
    }
}

// ---------------------------------------------------------------- scalar outputs
__global__ void k_scalar_combine(const float* scores, const float* h0, const float* f0,
                                 const float* f1, const float* meta,
                                 float* out_sc, float* out_h0c, float* out_mask) {
    int i = blockIdx.x * blockDim.x + threadIdx.x;
    if (i >= N_NODE) return;
    float fac = meta[0] * f0[i] + meta[1] * f1[i] + meta[2];
    float sc = fac * scores[i];
    out_sc[i] = sc;
    out_mask[i] = (sc != 0.0f) ? 1.0f : 0.0f;
    const float4* src = (const float4*)(h0 + (size_t)i * HID);
    float4* dst = (float4*)(out_h0c + (size_t)i * HID);
    #pragma unroll
    for (int j = 0; j < HID / 4; ++j) {
        float4 v = src[j];
        v.x *= fac; v.y *= fac; v.z *= fac; v.w *= fac;
        dst[j] = v;
    }
}
__global__ void k_nodes_out(const int* nodes, float* out_nodes) {
    int i = blockIdx.x * blockDim.x + threadIdx.x;
    if (i < 2 * N_NODE) out_nodes[i] = (float)nodes[i];
}

// ================================================================ launch
extern "C" void kernel_launch(void* const* d_in, const int* in_sizes, int n_in,
                              void* d_out, int out_size, void* d_ws, size_t ws_size,
                              hipStream_t stream) {
    const float* hidden  = (const float*)d_in[0];
    const float* scores  = (const float*)d_in[1];
    const float* h0      = (const float*)d_in[2];
    const float* alpha   = (const float*)d_in[3];
    const float* message = (const float*)d_in[4];
    const float* noise   = (const float*)d_in[5];
    const float* W1      = (const float*)d_in[6];
    const float* b1      = (const float*)d_in[7];
    const float* W2      = (const float*)d_in[8];
    const float* b2      = (const float*)d_in[9];
    const float* sub_tab = (const float*)d_in[10];
    const float* rel_tab = (const float*)d_in[11];
    const float* W_alpha = (const float*)d_in[12];
    const float* Wn      = (const float*)d_in[13];
    const int*   nodes   = (const int*)d_in[14];
    const int*   q_sub   = (const int*)d_in[15];
    const int*   q_rel   = (const int*)d_in[16];
    const int*   edges   = (const int*)d_in[17];
    const int*   old_idx = (const int*)d_in[18];
    const int*   l_ptr   = (const int*)d_in[19];

    float* out       = (float*)d_out;
    float* out_hc    = out;                       // 19,200,000
    float* out_nodes = out + 19200000;            //    600,000
    float* out_sc    = out + 19800000;            //    300,000
    float* out_h0c   = out + 20100000;            // 19,200,000
    float* out_mask  = out + 39300000;            //    300,000

    char* p = (char*)d_ws;
    auto bump = [&](size_t bytes) -> char* {
        char* r = p;
        p += (bytes + 255) & ~(size_t)255;
        return r;
    };
    float*        meta      = (float*)bump(256);
    float*        partials  = (float*)bump(NPART * 4);
    unsigned*     hist      = (unsigned*)bump(2048 * 4);
    int*          row_count = (int*)bump(64 * 4);
    float*        rowstats  = (float*)bump(64 * 4 * 4);
    int*          cand_node = (int*)bump((size_t)64 * ROWCAP * 4);
    int*          cand_col  = (int*)bump((size_t)64 * ROWCAP * 4);
    float*        cand_s    = (float*)bump((size_t)64 * ROWCAP * 4);
    float*        cand_q    = (float*)bump((size_t)64 * ROWCAP * 4);
    float*        f0        = (float*)bump((size_t)N_NODE * 4);
    float*        f1        = (float*)bump((size_t)N_NODE * 4);
    unsigned char* isnew    = (unsigned char*)bump(N_NODE);
    float*        ep        = (float*)bump((size_t)N_EDGE * 4);
    float*        ae        = (float*)bump((size_t)N_EDGE * 4);
    int*          deg       = (int*)bump((size_t)N_NODE * 4);
    int*          off       = (int*)bump((size_t)(N_NODE + 1) * 4);
    int*          cursor    = (int*)bump((size_t)N_NODE * 4);
    int*          elist     = (int*)bump((size_t)N_EDGE * 4);
    int*          eqlist    = (int*)bump((size_t)CAPQ * 4);
    float*        agg       = (float*)bump((size_t)N_NODE * HID * 4);

    const int TB = 256;
    int gN  = (N_NODE + TB - 1) / TB;
    int gO  = (N_OLD + TB - 1) / TB;
    int gE  = (N_EDGE + TB - 1) / TB;
    int gR  = (64 * ROWCAP + TB - 1) / TB;
    int g2N = (2 * N_NODE + TB - 1) / TB;

    k_init<<<gN, TB, 0, stream>>>(f0, f1, isnew, deg, cursor, row_count, meta);
    k_mark_old<<<gO, TB, 0, stream>>>(isnew, old_idx);
    k_gating<<<1, TB, 0, stream>>>(sub_tab, rel_tab, W1, b1, W2, b2, Wn, noise,
                                   q_sub, q_rel, l_ptr, meta);
    k_cand<<<gN, TB, 0, stream>>>(nodes, isnew, scores, hidden, rel_tab, q_rel,
                                  row_count, cand_node, cand_col, cand_s, cand_q);
    k_rowstats<<<64, TB, 0, stream>>>(row_count, cand_s, cand_q, rowstats);
    k_rank<<<gR, TB, 0, stream>>>(row_count, cand_node, cand_col, cand_s, cand_q,
                                  rowstats, meta, f0, f1);

    k_alpha_partmax<<<NPART, TB, 0, stream>>>(alpha, partials);
    k_alpha_finmax<<<1, TB, 0, stream>>>(partials, meta);
    k_alpha_partsum<<<NPART, TB, 0, stream>>>(alpha, meta, partials);
    k_alpha_finsum<<<1, TB, 0, stream>>>(partials, meta);
    k_ep<<<gE, TB, 0, stream>>>(alpha, meta, ep);

    // radix select: levels (shift, width, hi_shift, nbins)
    const int lv_s[3]  = {21, 10, 0};
    const int lv_w[3]  = {11, 11, 10};
    const int lv_hs[3] = {32, 21, 10};
    for (int lv = 0; lv < 3; ++lv) {
        k_hist_zero<<<8, TB, 0, stream>>>(hist, meta, lv == 0 ? 1 : 0);
        k_hist<<<gE, TB, 0, stream>>>(ep, edges, isnew, hist, meta,
                                      lv_s[lv], lv_hs[lv], 1 << lv_w[lv]);
        k_scan<<<1, 1, 0, stream>>>(hist, meta, lv_w[lv], 1 << lv_w[lv]);
    }
    k_eq_compact<<<gE, TB, 0, stream>>>(ep, edges, isnew, meta, eqlist);
    k_abase<<<gE, TB, 0, stream>>>(ep, alpha, edges, isnew, meta, ae);
    k_tiefix<<<(CAPQ + TB - 1) / TB, TB, 0, stream>>>(ep, alpha, meta, eqlist, ae);

    k_csr_count<<<gE, TB, 0, stream>>>(edges, deg);
    k_scan_off<<<1, 1024, 0, stream>>>(deg, off);
    k_csr_fill<<<gE, TB, 0, stream>>>(edges, off, cursor, elist);
    k_agg<<<(N_NODE * 8 + TB - 1) / TB, TB, 0, stream>>>(deg, off, elist, ae, message, agg);

    // 300000/16 * 64/16 = 75000 tiles, 8 waves per block
    k_gemm_combine<<<75000 / 8, TB, 0, stream>>>(agg, W_alpha, hidden, f0, f1, meta, out_hc);

    k_scalar_combine<<<gN, TB, 0, stream>>>(scores, h0, f0, f1, meta,
                                            out_sc, out_h0c, out_mask);
    k_nodes_out<<<g2N, TB, 0, stream>>>(nodes, out_nodes);
}